// VectorQuantizer_7481833029767
// MI455X (gfx1250) — compile-verified
//
#include <hip/hip_runtime.h>
#include <hip/hip_bf16.h>
#include <float.h>
#include <math.h>

typedef float v2f __attribute__((ext_vector_type(2)));
typedef float v8f __attribute__((ext_vector_type(8)));
typedef int   v4i __attribute__((ext_vector_type(4)));

#define C_DIM 256
#define HW    4096          // H*W
#define N_E   1024
#define N_TOT 65536         // B*H*W = 16*64*64
#define OUT_COS_N   262144  // B*W*C
#define OUT_IDX_N   65536
#define ROWS_BLK    128     // rows per block (8 waves x 16)
#define CODES_CHK   64      // codes per LDS chunk
#define B_STRIDE    260     // padded row stride (floats): conflict-free ds_load_b64
#define NCHUNK      (N_E / CODES_CHK)

// ---------------- CDNA5 async global->LDS helpers ----------------
__device__ __forceinline__ void async_copy_b128(const float* g, float* l) {
#if __has_builtin(__builtin_amdgcn_global_load_async_to_lds_b128)
    typedef __attribute__((address_space(1))) v4i* gp_t;   // global src
    typedef __attribute__((address_space(3))) v4i* lp_t;   // LDS dst
    __builtin_amdgcn_global_load_async_to_lds_b128((gp_t)g, (lp_t)l, 0, 0);
#else
    // VGLOBAL GV mode: vdst = LDS byte address, v[adr:adr+1] = 64-bit global addr
    unsigned lds = (unsigned)(unsigned long long)l;   // generic AS3 addr[31:0] = LDS offset
    asm volatile("global_load_async_to_lds_b128 %0, %1, off"
                 :: "v"(lds), "v"(g) : "memory");
#endif
}

__device__ __forceinline__ void wait_async0() {
#if __has_builtin(__builtin_amdgcn_s_wait_asynccnt)
    __builtin_amdgcn_s_wait_asynccnt(0);
#else
    asm volatile("s_wait_asynccnt 0" ::: "memory");
#endif
}

// ---------------- codebook squared norms ----------------
__global__ void cnorm_kernel(const float* __restrict__ cb, float* __restrict__ cnorm) {
    int j = blockIdx.x * blockDim.x + threadIdx.x;
    if (j >= N_E) return;
    const float4* row = (const float4*)(cb + (size_t)j * C_DIM);
    float acc = 0.f;
#pragma unroll 8
    for (int k = 0; k < C_DIM / 4; ++k) {
        float4 v = row[k];
        acc += v.x * v.x + v.y * v.y + v.z * v.z + v.w * v.w;
    }
    cnorm[j] = acc;
}

// ---------------- nearest-code argmin via f32 WMMA ----------------
// x layout: (B, C, H, W). Flattened row n=(b,h,w): x[b*C*HW + c*HW + (h*W+w)].
// Block: 256 threads (8 waves), 128 rows. Codebook chunks (64 codes) stream
// through a double-buffered LDS tile filled by async global->LDS copies.
__global__ __launch_bounds__(256)
void argmin_kernel(const float* __restrict__ x,
                   const float* __restrict__ cb,
                   const float* __restrict__ cnorm,
                   int*   __restrict__ idx_out,
                   float* __restrict__ fidx_out) {
    // A: [kpair 0..127][row 0..127], element = (K=2p, K=2p+1)  -> 128 KB
    __shared__ v2f   A2[128 * ROWS_BLK];
    // B: two buffers, row-major padded: code j, dim k at Bst[buf][j*260 + k]
    __shared__ float Bst[2][CODES_CHK * B_STRIDE];   // 2 x 65 KB

    const int tid  = threadIdx.x;
    const int lane = tid & 31;
    const int wave = tid >> 5;
    const int m    = lane & 15;    // M (A) / N (B,C,D) position
    const int hk   = lane >> 4;    // which K-half of the 16x16x4 fragment
    const int r0   = wave * 16;

    const int n0  = blockIdx.x * ROWS_BLK;
    const int b   = n0 / HW;
    const int hw0 = n0 % HW;       // 128 consecutive rows never cross b (4096 % 128 == 0)
    const float* xb = x + (size_t)b * C_DIM * HW + hw0;

    // Stage A: 128 rows x 256 dims, coalesced global reads, ds_store_b64 writes.
    for (int i = tid; i < 128 * ROWS_BLK; i += 256) {
        int r = i & (ROWS_BLK - 1);
        int p = i >> 7;
        v2f v;
        v.x = xb[(size_t)(2 * p    ) * HW + r];
        v.y = xb[(size_t)(2 * p + 1) * HW + r];
        A2[p * ROWS_BLK + r] = v;
    }

    // Kick off async fill of chunk 0 into buffer 0.
#pragma unroll 4
    for (int u = tid; u < CODES_CHK * 64; u += 256) {
        int j = u >> 6;          // code within chunk
        int q = u & 63;          // 16-byte quad within the 1KB code row
        async_copy_b128(cb + (size_t)j * C_DIM + q * 4, &Bst[0][j * B_STRIDE + q * 4]);
    }
    wait_async0();
    __syncthreads();             // A2 + chunk 0 visible to all waves

    float bestv[8];
    int   besti[8];
#pragma unroll
    for (int r = 0; r < 8; ++r) { bestv[r] = FLT_MAX; besti[r] = 0; }

    for (int c = 0; c < NCHUNK; ++c) {
        const int    cbase = c * CODES_CHK;
        const float* Bcur  = Bst[c & 1];

        if (c + 1 < NCHUNK) {
            // Async-fill next chunk while this one computes.
            const float* gsrc = cb + (size_t)(cbase + CODES_CHK) * C_DIM;
            float*       ldst = Bst[(c + 1) & 1];
#pragma unroll 4
            for (int u = tid; u < CODES_CHK * 64; u += 256) {
                int j = u >> 6;
                int q = u & 63;
                async_copy_b128(gsrc + (size_t)j * C_DIM + q * 4, ldst + j * B_STRIDE + q * 4);
            }
            // Warm L2 two chunks ahead (256B per lane covers the 64KB chunk).
            if (c + 2 < NCHUNK)
                __builtin_prefetch(cb + (size_t)(cbase + 2 * CODES_CHK) * C_DIM + tid * 64, 0, 0);
        }

        v8f acc[4];
#pragma unroll
        for (int nt = 0; nt < 4; ++nt) acc[nt] = (v8f)0.0f;

#pragma unroll 4
        for (int kc = 0; kc < 64; ++kc) {
            int k = 4 * kc + 2 * hk;                      // fragment K (this lane half)
            v2f a  = A2[(2 * kc + hk) * ROWS_BLK + r0 + m];
            v2f b0 = *(const v2f*)&Bcur[(m     ) * B_STRIDE + k];
            v2f b1 = *(const v2f*)&Bcur[(m + 16) * B_STRIDE + k];
            v2f b2 = *(const v2f*)&Bcur[(m + 32) * B_STRIDE + k];
            v2f b3 = *(const v2f*)&Bcur[(m + 48) * B_STRIDE + k];
            acc[0] = __builtin_amdgcn_wmma_f32_16x16x4_f32(false, a, false, b0, (short)0, acc[0], false, false);
            acc[1] = __builtin_amdgcn_wmma_f32_16x16x4_f32(false, a, false, b1, (short)0, acc[1], false, false);
            acc[2] = __builtin_amdgcn_wmma_f32_16x16x4_f32(false, a, false, b2, (short)0, acc[2], false, false);
            acc[3] = __builtin_amdgcn_wmma_f32_16x16x4_f32(false, a, false, b3, (short)0, acc[3], false, false);
        }

        // score = ||c||^2 - 2*dot  (||x||^2 constant per row, irrelevant to argmin)
#pragma unroll
        for (int nt = 0; nt < 4; ++nt) {
            int   cid = cbase + nt * 16 + m;
            float cn  = cnorm[cid];
#pragma unroll
            for (int r = 0; r < 8; ++r) {
                float s = cn - 2.0f * acc[nt][r];
                if (s < bestv[r]) { bestv[r] = s; besti[r] = cid; }
            }
        }

        if (c + 1 < NCHUNK) wait_async0();   // next buffer filled (own wave's ops)
        __syncthreads();                     // everyone done reading Bcur; next buf visible
    }

    // Cross-lane argmin: distances for row M live across the 16 lanes of one
    // C-slot; xor offsets 1..8 keep the two 16-lane halves separate.
#pragma unroll
    for (int r = 0; r < 8; ++r) {
        float bv = bestv[r];
        int   bi = besti[r];
#pragma unroll
        for (int off = 8; off >= 1; off >>= 1) {
            float ov = __shfl_xor(bv, off, 32);
            int   oi = __shfl_xor(bi, off, 32);
            if (ov < bv || (ov == bv && oi < bi)) { bv = ov; bi = oi; }
        }
        if (m == 0) {
            int row = n0 + r0 + r + hk * 8;   // lanes 0 / 16 hold M=r / M=8+r
            idx_out[row]  = bi;
            fidx_out[row] = (float)bi;
        }
    }
}

// ---------------- gather + cosine over H ----------------
// out[(b*W + w)*C + c] ; block = one (b,w), 256 lanes walk c.
__global__ __launch_bounds__(256)
void cosine_kernel(const float* __restrict__ cb,
                   const int* __restrict__ idx_z,
                   const int* __restrict__ idx_gt,
                   float* __restrict__ out) {
    __shared__ int s_iz[64];
    __shared__ int s_ig[64];
    const int bw = blockIdx.x;        // b*W + w
    const int b  = bw >> 6;
    const int w  = bw & 63;
    const int c  = threadIdx.x;

    if (c < 64) {
        s_iz[c] = idx_z[(b * 64 + c) * 64 + w];
    } else if (c < 128) {
        int h = c - 64;
        s_ig[h] = idx_gt[(b * 64 + h) * 64 + w];
    }
    __syncthreads();

    float num = 0.f, xx = 0.f, yy = 0.f;
#pragma unroll 4
    for (int h = 0; h < 64; ++h) {
        float a = cb[(size_t)s_ig[h] * C_DIM + c];   // z_q_gt
        float q = cb[(size_t)s_iz[h] * C_DIM + c];   // z_q
        num += a * q;
        xx  += a * a;
        yy  += q * q;
    }
    float nx = fmaxf(sqrtf(xx), 1e-8f);
    float ny = fmaxf(sqrtf(yy), 1e-8f);
    out[(size_t)bw * C_DIM + c] = num / (nx * ny);
}

extern "C" void kernel_launch(void* const* d_in, const int* in_sizes, int n_in,
                              void* d_out, int out_size, void* d_ws, size_t ws_size,
                              hipStream_t stream) {
    const float* z  = (const float*)d_in[0];
    const float* gt = (const float*)d_in[1];
    const float* cb = (const float*)d_in[2];
    float* out = (float*)d_out;

    float* cnorm  = (float*)d_ws;                                   // 1024 f
    int*   idx_z  = (int*)((char*)d_ws + 4096);                     // 65536 i32
    int*   idx_gt = (int*)((char*)d_ws + 4096 + (size_t)N_TOT * 4); // 65536 i32

    // d_out layout (reference tuple order): [cosine | idx_gt | idx]
    float* out_cos    = out;
    float* out_idx_gt = out + OUT_COS_N;
    float* out_idx    = out + OUT_COS_N + OUT_IDX_N;

    cnorm_kernel<<<4, 256, 0, stream>>>(cb, cnorm);
    argmin_kernel<<<N_TOT / ROWS_BLK, 256, 0, stream>>>(z,  cb, cnorm, idx_z,  out_idx);
    argmin_kernel<<<N_TOT / ROWS_BLK, 256, 0, stream>>>(gt, cb, cnorm, idx_gt, out_idx_gt);
    cosine_kernel<<<16 * 64, 256, 0, stream>>>(cb, idx_z, idx_gt, out_cos);
}